// StructuralFeatureEncoder_76347338654282
// MI455X (gfx1250) — compile-verified
//
#include <hip/hip_runtime.h>
#include <hip/hip_bf16.h>

typedef float v2f __attribute__((ext_vector_type(2)));
typedef float v8f __attribute__((ext_vector_type(8)));

#define D 128

// ---------------- utility ----------------
__global__ void zero_kernel(float* __restrict__ p, int n) {
    int i = blockIdx.x * blockDim.x + threadIdx.x;
    if (i < n) p[i] = 0.0f;
}

// ---------------- degree / normalization ----------------
__global__ void deg_kernel(const int* __restrict__ dst, float* __restrict__ deg, int E) {
    int e = blockIdx.x * blockDim.x + threadIdx.x;
    if (e < E) unsafeAtomicAdd(&deg[dst[e]], 1.0f);
}

__global__ void dinv_kernel(const float* __restrict__ deg, float* __restrict__ dinv, int n) {
    int i = blockIdx.x * blockDim.x + threadIdx.x;
    if (i < n) dinv[i] = rsqrtf(deg[i] + 1.0f);   // +1 for self loop
}

// ---------------- GEMM: H = X @ W  via f32 WMMA 16x16x4 ----------------
// One wave per 16x16 output tile. Block = 256 threads = 8 waves = one row
// tile across all 8 column tiles (D=128 -> 8 tiles of 16 cols).
__global__ void gemm_wmma_kernel(const float* __restrict__ X,
                                 const float* __restrict__ W,
                                 float* __restrict__ H, int nRows) {
    int wave = (blockIdx.x * blockDim.x + threadIdx.x) >> 5;
    int lane = threadIdx.x & 31;
    int rowTile = wave >> 3;      // 8 col tiles per row tile
    int colTile = wave & 7;
    int row0 = rowTile * 16;
    if (row0 >= nRows) return;
    int col0 = colTile * 16;

    int mrow  = row0 + (lane & 15);   // A: lane holds row M=lane%16
    int khalf = lane >> 4;            // A/B: lanes 16-31 hold K+2,K+3
    int ncol  = col0 + (lane & 15);   // B: lane holds col N=lane%16

    v8f c = {};
    #pragma unroll 4
    for (int k = 0; k < D; k += 4) {
        int ka = k + khalf * 2;
        v2f a, bm;
        a.x  = X[(size_t)mrow * D + ka];
        a.y  = X[(size_t)mrow * D + ka + 1];
        bm.x = W[(size_t)(ka)     * D + ncol];
        bm.y = W[(size_t)(ka + 1) * D + ncol];
        c = __builtin_amdgcn_wmma_f32_16x16x4_f32(
                /*neg_a=*/false, a, /*neg_b=*/false, bm,
                /*c_mod=*/(short)0, c, /*reuse_a=*/false, /*reuse_b=*/false);
    }

    // C layout: VGPR r -> row = row0 + r + (lane/16)*8, col = col0 + lane%16
    int crow = row0 + ((lane >> 4) << 3);
    int ccol = col0 + (lane & 15);
    #pragma unroll
    for (int r = 0; r < 8; ++r)
        H[(size_t)(crow + r) * D + ccol] = c[r];
}

// ---------------- self-loop init: agg = h * dinv^2  (full write) ----------------
__global__ void selfloop_kernel(const float* __restrict__ h,
                                const float* __restrict__ dinv,
                                float* __restrict__ agg, int n) {
    int t = blockIdx.x * blockDim.x + threadIdx.x;   // n*32 threads
    int node = t >> 5, lane = t & 31;
    if (node >= n) return;
    float di = dinv[node];
    float cc = di * di;
    float4 v = ((const float4*)(h + (size_t)node * D))[lane];
    float4 o = make_float4(v.x * cc, v.y * cc, v.z * cc, v.w * cc);
    ((float4*)(agg + (size_t)node * D))[lane] = o;
}

// ---------------- edge scatter: agg[dst] += coef * h[src] ----------------
// One wave per edge: 32 lanes x float4 = 128 floats, coalesced 512B gather,
// hardware fp32 atomics into L2 (node array fits entirely in 192MB L2).
__global__ void edge_kernel(const int* __restrict__ src,
                            const int* __restrict__ dst,
                            const float* __restrict__ dinv,
                            const float* __restrict__ h,
                            float* __restrict__ agg, int E) {
    int t = blockIdx.x * blockDim.x + threadIdx.x;   // E*32 threads
    int e = t >> 5, lane = t & 31;
    if (e >= E) return;
    int s = src[e], d = dst[e];
    float coef = dinv[s] * dinv[d];
    float4 v = ((const float4*)(h + (size_t)s * D))[lane];
    float* ap = agg + (size_t)d * D + lane * 4;
    unsafeAtomicAdd(ap + 0, v.x * coef);
    unsafeAtomicAdd(ap + 1, v.y * coef);
    unsafeAtomicAdd(ap + 2, v.z * coef);
    unsafeAtomicAdd(ap + 3, v.w * coef);
}

// ---------------- fused residual + bias + LayerNorm + ReLU ----------------
// One wave per node; wave32 shuffle reduction for mean/var.
__global__ void ln_relu_kernel(const float* __restrict__ xcur,
                               const float* __restrict__ agg,
                               const float* __restrict__ bias,
                               const float* __restrict__ gamma,
                               const float* __restrict__ beta,
                               float* __restrict__ xnext, int n) {
    int t = blockIdx.x * blockDim.x + threadIdx.x;   // n*32 threads
    int node = t >> 5, lane = t & 31;
    if (node >= n) return;
    float4 x = ((const float4*)(xcur + (size_t)node * D))[lane];
    float4 a = ((const float4*)(agg  + (size_t)node * D))[lane];
    float4 b = ((const float4*)bias)[lane];
    float4 v = make_float4(x.x + a.x + b.x, x.y + a.y + b.y,
                           x.z + a.z + b.z, x.w + a.w + b.w);
    float s  = v.x + v.y + v.z + v.w;
    float sq = v.x * v.x + v.y * v.y + v.z * v.z + v.w * v.w;
    #pragma unroll
    for (int off = 16; off > 0; off >>= 1) {
        s  += __shfl_xor(s,  off, 32);
        sq += __shfl_xor(sq, off, 32);
    }
    float mean = s * (1.0f / 128.0f);
    float var  = sq * (1.0f / 128.0f) - mean * mean;
    float inv  = rsqrtf(var + 1e-5f);
    float4 g  = ((const float4*)gamma)[lane];
    float4 bt = ((const float4*)beta)[lane];
    float4 y;
    y.x = fmaxf((v.x - mean) * inv * g.x + bt.x, 0.0f);
    y.y = fmaxf((v.y - mean) * inv * g.y + bt.y, 0.0f);
    y.z = fmaxf((v.z - mean) * inv * g.z + bt.z, 0.0f);
    y.w = fmaxf((v.w - mean) * inv * g.w + bt.w, 0.0f);
    ((float4*)(xnext + (size_t)node * D))[lane] = y;
}

// ---------------- global mean pool ----------------
__global__ void pool_kernel(const float* __restrict__ x,
                            const int* __restrict__ batch,
                            float* __restrict__ pool,
                            float* __restrict__ cnt, int n) {
    int t = blockIdx.x * blockDim.x + threadIdx.x;   // n*32 threads
    int node = t >> 5, lane = t & 31;
    if (node >= n) return;
    int g = batch[node];
    float4 v = ((const float4*)(x + (size_t)node * D))[lane];
    float* pp = pool + (size_t)g * D + lane * 4;
    unsafeAtomicAdd(pp + 0, v.x);
    unsafeAtomicAdd(pp + 1, v.y);
    unsafeAtomicAdd(pp + 2, v.z);
    unsafeAtomicAdd(pp + 3, v.w);
    if (lane == 0) unsafeAtomicAdd(&cnt[g], 1.0f);
}

__global__ void finalize_kernel(const float* __restrict__ pool,
                                const float* __restrict__ cnt,
                                float* __restrict__ out, int total) {
    int i = blockIdx.x * blockDim.x + threadIdx.x;
    if (i < total) {
        int g = i >> 7;   // /128
        out[i] = pool[i] / fmaxf(cnt[g], 1.0f);
    }
}

// ---------------- host orchestration ----------------
extern "C" void kernel_launch(void* const* d_in, const int* in_sizes, int n_in,
                              void* d_out, int out_size, void* d_ws, size_t ws_size,
                              hipStream_t stream) {
    const float* x     = (const float*)d_in[0];
    const int*   ei    = (const int*)  d_in[1];
    const int*   batch = (const int*)  d_in[2];
    const float* W     = (const float*)d_in[3];
    const float* b     = (const float*)d_in[4];
    const float* gamma = (const float*)d_in[5];
    const float* beta  = (const float*)d_in[6];
    float* out = (float*)d_out;

    const int N = in_sizes[0] / D;
    const int E = in_sizes[1] / 2;
    const int L = in_sizes[4] / D;
    const int G = out_size / D;
    const int* src = ei;
    const int* dst = ei + E;

    // workspace carve-up (floats)
    size_t nd = (size_t)N * D;
    float* h    = (float*)d_ws;
    float* bufA = h    + nd;
    float* bufB = bufA + nd;
    float* dinv = bufB + nd;
    float* deg  = dinv + N;
    float* pool = deg  + N;
    float* cnt  = pool + (size_t)G * D;   // pool & cnt contiguous

    const int T = 256;
    int nw = N * 32;   // one wave per node
    int ew_blocks = (int)(((size_t)E * 32 + T - 1) / T);

    // degree + normalization (depends only on dst; once per call)
    zero_kernel<<<(N + T - 1) / T, T, 0, stream>>>(deg, N);
    deg_kernel <<<(E + T - 1) / T, T, 0, stream>>>(dst, deg, E);
    dinv_kernel<<<(N + T - 1) / T, T, 0, stream>>>(deg, dinv, N);

    const float* xcur = x;
    for (int l = 0; l < L; ++l) {
        float* agg = (l & 1) ? bufB : bufA;
        int rowTiles = (N + 15) / 16;
        gemm_wmma_kernel<<<rowTiles, T, 0, stream>>>(xcur, W + (size_t)l * D * D, h, N);
        selfloop_kernel <<<(nw + T - 1) / T, T, 0, stream>>>(h, dinv, agg, N);
        edge_kernel     <<<ew_blocks, T, 0, stream>>>(src, dst, dinv, h, agg, E);
        // in-place on agg: each thread reads then writes its own float4
        ln_relu_kernel  <<<(nw + T - 1) / T, T, 0, stream>>>(
            xcur, agg, b + (size_t)l * D, gamma + (size_t)l * D,
            beta + (size_t)l * D, agg, N);
        xcur = agg;
    }

    // global mean pool
    zero_kernel    <<<(G * D + G + T - 1) / T, T, 0, stream>>>(pool, G * D + G);
    pool_kernel    <<<(nw + T - 1) / T, T, 0, stream>>>(xcur, batch, pool, cnt, N);
    finalize_kernel<<<(G * D + T - 1) / T, T, 0, stream>>>(pool, cnt, out, G * D);
}